// BurgersParamPDE_85478439125431
// MI455X (gfx1250) — compile-verified
//
#include <hip/hip_runtime.h>
#include <hip/hip_bf16.h>

// Burgers upwind stencil, N = 2^25 fp32.
// Bandwidth-bound: min traffic = 8 B/element (4 read + 4 write) ~= 268 MB
// -> ~11.5 us at 23.3 TB/s HBM. No matmul structure, so WMMA is inapplicable;
// the MI455X-optimal shape is pure wide-vector streaming:
//   - 8 elements/thread: 2x global_load_b128 + 2 halo b32 loads (cache hits)
//   - 2x global_store_b128 with NT temporal hint (output never re-read;
//     input+output = 256 MB > 192 MB L2, keep the read stream resident)
//   - wave32-clean, 256-thread blocks.

typedef float v4f __attribute__((ext_vector_type(4)));

__device__ __forceinline__ float burgers_pt(float l, float c, float r, float d) {
    const float inv_dx  = 100.0f;    // fp32(1/0.01)
    const float inv_dx2 = 10000.0f;  // fp32((1/0.01)^2)
    float delta = (l - 2.0f * c + r) * inv_dx2;   // centered 2nd difference
    float advl  = (c - l) * inv_dx;               // backward difference
    float advr  = (r - c) * inv_dx;               // forward difference
    float adv   = fmaxf(c, 0.0f) * advl + fminf(c, 0.0f) * advr;  // upwind
    return d * delta - adv;
}

__global__ __launch_bounds__(256) void burgers_stencil_kernel(
    const float* __restrict__ u,       // [N] state
    const float* __restrict__ bc,      // [2] Dirichlet boundary values
    const float* __restrict__ d_org,   // [1] raw diffusion parameter
    float* __restrict__ out,           // [N]
    int n)
{
    const int  tid  = blockIdx.x * blockDim.x + threadIdx.x;
    const long base = (long)tid * 8;
    if (base >= n) return;

    // d = sigmoid(d_org) * 0.01  (uniform; one exp per thread, negligible)
    const float d = 0.01f / (1.0f + expf(-d_org[0]));

    if (base + 8 <= n) {
        // Interior fast path: two 128-bit vector loads own the 8 center
        // values; the two halo scalars land on cachelines already pulled in
        // by neighboring lanes (no extra HBM traffic).
        const v4f c0 = *reinterpret_cast<const v4f*>(u + base);
        const v4f c1 = *reinterpret_cast<const v4f*>(u + base + 4);
        const float left  = (base == 0)     ? bc[0] : u[base - 1];
        const float right = (base + 8 == n) ? bc[1] : u[base + 8];

        v4f o0, o1;
        o0.x = burgers_pt(left, c0.x, c0.y, d);
        o0.y = burgers_pt(c0.x, c0.y, c0.z, d);
        o0.z = burgers_pt(c0.y, c0.z, c0.w, d);
        o0.w = burgers_pt(c0.z, c0.w, c1.x, d);
        o1.x = burgers_pt(c0.w, c1.x, c1.y, d);
        o1.y = burgers_pt(c1.x, c1.y, c1.z, d);
        o1.z = burgers_pt(c1.y, c1.z, c1.w, d);
        o1.w = burgers_pt(c1.z, c1.w, right, d);

        // NT stores: output is write-once; don't let 128 MB of it evict the
        // input stream from the 192 MB L2.
        __builtin_nontemporal_store(o0, reinterpret_cast<v4f*>(out + base));
        __builtin_nontemporal_store(o1, reinterpret_cast<v4f*>(out + base + 4));
    } else {
        // Ragged tail (not hit for N = 2^25, kept for generality).
        for (long i = base; i < n; ++i) {
            const float l = (i == 0)     ? bc[0] : u[i - 1];
            const float c = u[i];
            const float r = (i + 1 == n) ? bc[1] : u[i + 1];
            out[i] = burgers_pt(l, c, r, d);
        }
    }
}

extern "C" void kernel_launch(void* const* d_in, const int* in_sizes, int n_in,
                              void* d_out, int out_size, void* d_ws, size_t ws_size,
                              hipStream_t stream) {
    const float* u     = (const float*)d_in[0];  // state, [1,1,N] fp32
    const float* bc    = (const float*)d_in[1];  // [1,1,2] fp32
    const float* d_org = (const float*)d_in[2];  // scalar fp32
    float*       out   = (float*)d_out;

    const int n = in_sizes[0];                   // 2^25
    const int threads_total = (n + 7) / 8;       // 8 elements per thread
    const int block = 256;                       // 8 waves (wave32)
    const int grid  = (threads_total + block - 1) / block;

    burgers_stencil_kernel<<<grid, block, 0, stream>>>(u, bc, d_org, out, n);
}